// FieldTypeClassification_5634997092806
// MI455X (gfx1250) — compile-verified
//
#include <hip/hip_runtime.h>
#include <hip/hip_bf16.h>

// Problem constants (from reference)
#define BS   8
#define SEQ  512
#define C    512
#define NC   5
#define H    768
#define W    768
#define NTOK (BS * SEQ)      // 4096
#define HW   (H * W)         // 589824
#define LGST 8               // padded logits stride per token

typedef __attribute__((ext_vector_type(2))) float v2f;
typedef __attribute__((ext_vector_type(8))) float v8f;

// ---------------------------------------------------------------------------
// Kernel 1: logits[t, n] = fuse_embeddings[t, :] . Wc[n, :] + bc[n]
// fp32 WMMA 16x16x4: one wave owns a 16-token tile, K-loop over 512 in 4s.
// A (16x4, MxK) lane layout: lanes 0-15 -> K={k,k+1}, lanes 16-31 -> K={k+2,k+3}
// B (4x16, KxN) lane layout mirrors A (N = lane&15).
// D (16x16): VGPR r, lanes 0-15 -> M=r; lanes 16-31 -> M=8+r; N = lane&15.
//
// Columns N>=NC are never stored, and in a matmul column N of D depends only
// on column N of B, so lanes with l15>=NC just need a VALID address (clamped
// to row NC-1) -- no zeroing, no EXEC divergence, pure b64 loads.
// ---------------------------------------------------------------------------
__global__ void logits_wmma_f32(const float* __restrict__ A,    // [NTOK, C]
                                const float* __restrict__ Wc,   // [NC, C]
                                const float* __restrict__ bc,   // [NC]
                                float* __restrict__ logits)     // [NTOK, LGST]
{
    const int lane = threadIdx.x & 31;
    const int wave = threadIdx.x >> 5;
    const int tile = blockIdx.x * (blockDim.x >> 5) + wave;   // 0..255
    const int m0   = tile * 16;
    const int half = lane >> 4;      // 0 or 1 -> K sub-pair
    const int l15  = lane & 15;      // M row (A) / N col (B)

    const int  bn   = (l15 < NC) ? l15 : (NC - 1);            // clamped B row
    const float* arow = A  + (long)(m0 + l15) * C + 2 * half;
    const float* bcol = Wc + (long)bn * C + 2 * half;

    v8f acc = {};
    #pragma unroll 4
    for (int k = 0; k < C; k += 4) {
        const v2f a = *(const v2f*)(arow + k);   // global_load_b64
        const v2f b = *(const v2f*)(bcol + k);   // global_load_b64
        // (neg_a, A, neg_b, B, c_mod, C, reuse_a, reuse_b)
        acc = __builtin_amdgcn_wmma_f32_16x16x4_f32(
                  false, a, false, b, (short)0, acc, false, false);
    }

    if (l15 < NC) {
        const float bias = bc[l15];
        #pragma unroll
        for (int r = 0; r < 8; ++r) {
            const int M = r + 8 * half;
            logits[(long)(m0 + M) * LGST + l15] = acc[r] + bias;
        }
    }
}

// ---------------------------------------------------------------------------
// Kernel 2: per-pixel argmax over NC channels -> u8 label map (first-max tie).
// Streams the 94 MB class_labels once, coalesced; writes 4.7 MB.
// ---------------------------------------------------------------------------
__global__ void label_argmax_kernel(const float* __restrict__ cl,  // [BS,NC,H,W]
                                    unsigned char* __restrict__ lab) // [BS,H,W]
{
    const int idx = blockIdx.x * blockDim.x + threadIdx.x;
    if (idx >= BS * HW) return;
    const int b = idx / HW;
    const int p = idx - b * HW;
    const float* base = cl + (long)b * NC * HW + p;
    float best = base[0];
    int bi = 0;
    #pragma unroll
    for (int c = 1; c < NC; ++c) {
        const float v = base[(long)c * HW];
        if (v > best) { best = v; bi = c; }   // strict > => first-max tiebreak
    }
    lab[idx] = (unsigned char)bi;
}

// ---------------------------------------------------------------------------
// Kernel 3: one wave per token. Box majority vote + cross entropy.
// USE_LAB=true reads the u8 label map; false recomputes argmax per pixel.
// ---------------------------------------------------------------------------
template <bool USE_LAB>
__global__ void token_ce_kernel(const unsigned char* __restrict__ lab,  // [BS,H,W] (may be null)
                                const float* __restrict__ cl,           // [BS,NC,H,W]
                                const float* __restrict__ logits,       // [NTOK, LGST]
                                const int*   __restrict__ coords,       // [NTOK, 4]
                                const int*   __restrict__ mask,         // [NTOK]
                                float* __restrict__ nll_out,            // [NTOK]
                                float* __restrict__ m_out)              // [NTOK]
{
    const int lane = threadIdx.x & 31;
    const int t = blockIdx.x * (blockDim.x >> 5) + (threadIdx.x >> 5);
    if (t >= NTOK) return;   // whole-wave exit (tokens map to whole waves)
    const int b = t / SEQ;

    const int4 cd = ((const int4*)coords)[t];   // x0, y0, x1, y1
    int x0 = cd.x, y0 = cd.y, x1 = cd.z, y1 = cd.w;
    if (y1 == y0) y1 += 1;                      // degenerate-box fix
    if (x1 == x0) x1 += 1;
    const int w = x1 - x0;
    const int area = w * (y1 - y0);

    int cnt[NC] = {0, 0, 0, 0, 0};
    if (USE_LAB) {
        const unsigned char* lb = lab + (long)b * HW;
        for (int i = lane; i < area; i += 32) {
            const int yy = y0 + i / w;
            const int xx = x0 + i - (i / w) * w;
            cnt[lb[yy * W + xx]]++;
        }
    } else {
        const float* base = cl + (long)b * NC * HW;
        for (int i = lane; i < area; i += 32) {
            const int yy = y0 + i / w;
            const int xx = x0 + i - (i / w) * w;
            const int p = yy * W + xx;
            float best = base[p];
            int bi = 0;
            #pragma unroll
            for (int c = 1; c < NC; ++c) {
                const float v = base[(long)c * HW + p];
                if (v > best) { best = v; bi = c; }
            }
            cnt[bi]++;
        }
    }

    // wave32 reduction of the 5 counters
    #pragma unroll
    for (int c = 0; c < NC; ++c) {
        #pragma unroll
        for (int off = 16; off > 0; off >>= 1)
            cnt[c] += __shfl_xor(cnt[c], off, 32);
    }

    if (lane == 0) {
        int maj = 0, bestc = cnt[0];
        #pragma unroll
        for (int c = 1; c < NC; ++c)
            if (cnt[c] > bestc) { bestc = cnt[c]; maj = c; }  // first-max tie

        const float* lg = logits + (long)t * LGST;
        float mx = lg[0];
        #pragma unroll
        for (int c = 1; c < NC; ++c) mx = fmaxf(mx, lg[c]);
        float se = 0.0f;
        #pragma unroll
        for (int c = 0; c < NC; ++c) se += expf(lg[c] - mx);
        const float nll = (mx + logf(se)) - lg[maj];
        const float m = (mask[t] == 1) ? 1.0f : 0.0f;
        nll_out[t] = nll * m;
        m_out[t]   = m;
    }
}

// ---------------------------------------------------------------------------
// Kernel 4: deterministic fixed-order reduction -> scalar mean.
// ---------------------------------------------------------------------------
__global__ void reduce_final_kernel(const float* __restrict__ nll,
                                    const float* __restrict__ m,
                                    float* __restrict__ out)
{
    __shared__ float s_n[256];
    __shared__ float s_m[256];
    const int tid = threadIdx.x;
    float sn = 0.0f, sm = 0.0f;
    for (int i = tid; i < NTOK; i += 256) { sn += nll[i]; sm += m[i]; }
    s_n[tid] = sn; s_m[tid] = sm;
    __syncthreads();
    for (int off = 128; off > 0; off >>= 1) {
        if (tid < off) { s_n[tid] += s_n[tid + off]; s_m[tid] += s_m[tid + off]; }
        __syncthreads();
    }
    if (tid == 0) out[0] = s_n[0] / s_m[0];
}

// ---------------------------------------------------------------------------
extern "C" void kernel_launch(void* const* d_in, const int* in_sizes, int n_in,
                              void* d_out, int out_size, void* d_ws, size_t ws_size,
                              hipStream_t stream)
{
    const float* fuse   = (const float*)d_in[0];   // [BS, SEQ, C]
    const float* cl     = (const float*)d_in[1];   // [BS, NC, H, W]
    const float* Wc     = (const float*)d_in[2];   // [NC, C]
    const float* bc     = (const float*)d_in[3];   // [NC]
    const int*   coords = (const int*)  d_in[4];   // [BS, SEQ, 4]
    const int*   mask   = (const int*)  d_in[5];   // [BS, SEQ]
    float* out = (float*)d_out;

    // workspace layout
    char* ws = (char*)d_ws;
    const size_t off_logits = 0;
    const size_t off_nll    = off_logits + (size_t)NTOK * LGST * sizeof(float); // 128 KB
    const size_t off_m      = off_nll + (size_t)NTOK * sizeof(float);
    const size_t off_lab    = off_m + (size_t)NTOK * sizeof(float);
    const size_t need_lab   = off_lab + (size_t)BS * HW;                        // ~4.9 MB

    float* logits = (float*)(ws + off_logits);
    float* nllbuf = (float*)(ws + off_nll);
    float* mbuf   = (float*)(ws + off_m);
    unsigned char* lab = (unsigned char*)(ws + off_lab);
    const bool use_lab = (ws_size >= need_lab);

    // 1) WMMA GEMM: 256 tiles of 16 tokens; 8 waves/block -> 32 blocks.
    logits_wmma_f32<<<32, 256, 0, stream>>>(fuse, Wc, bc, logits);

    // 2) label map (if workspace allows)
    if (use_lab) {
        const int npix = BS * HW;
        label_argmax_kernel<<<(npix + 255) / 256, 256, 0, stream>>>(cl, lab);
    }

    // 3) per-token majority vote + CE: 4096 waves, 8 waves/block -> 512 blocks.
    if (use_lab) {
        token_ce_kernel<true><<<NTOK / 8, 256, 0, stream>>>(
            lab, cl, logits, coords, mask, nllbuf, mbuf);
    } else {
        token_ce_kernel<false><<<NTOK / 8, 256, 0, stream>>>(
            nullptr, cl, logits, coords, mask, nllbuf, mbuf);
    }

    // 4) deterministic reduction to the scalar output
    reduce_final_kernel<<<1, 256, 0, stream>>>(nllbuf, mbuf, out);
}